// LinearEncoder_14199161880910
// MI455X (gfx1250) — compile-verified
//
#include <hip/hip_runtime.h>

#define N_IN  128
#define N_OUT 64
// LDS pair-row stride: 160 floats == 32 (mod 64) banks -> lower/upper half-wave
// hit disjoint bank halves, conflict-free ds_load_b64.
#define WP_STRIDE 160

typedef __attribute__((ext_vector_type(2))) float v2f;
typedef __attribute__((ext_vector_type(8))) float v8f;

// ---------------------------------------------------------------- degree ----
__global__ void gcn_init_deg(float* __restrict__ deg, int n) {
    int i = blockIdx.x * blockDim.x + threadIdx.x;
    if (i < n) deg[i] = 1.0f;               // self-loop contributes 1 to degree
}

__global__ void gcn_deg_accum(const int* __restrict__ col, float* __restrict__ deg, int E) {
    int e = blockIdx.x * blockDim.x + threadIdx.x;
    if (e < E) atomicAdd(&deg[col[e]], 1.0f);
}

__global__ void gcn_rsqrt(float* __restrict__ deg, int n) {
    int i = blockIdx.x * blockDim.x + threadIdx.x;
    if (i < n) {
        float d = deg[i];
        deg[i] = (d > 0.0f) ? rsqrtf(d) : 0.0f;   // becomes dinv in place
    }
}

// ------------------------------------------------------------ WMMA GEMM -----
// h[N, 64] = x[N, 128] @ W[128, 64]
// One wave per 16-row block; 4 accumulators cover N=0..63.
// A 16x4 f32: lane L -> M = L%16, K = 2*(L/16)+v   (v2f per lane)
// B 4x16 f32: lane L -> N = L%16, K = 2*(L/16)+v   (v2f per lane)
// C 16x16 f32: lane L -> N = L%16, M = v + 8*(L/16) (v8f per lane)
// W is staged in LDS pre-packed so each lane's B fragment (W[k][n], W[k+1][n])
// is one contiguous b64: sW[(k/2)*WP_STRIDE + n*2 + (k&1)].
__global__ __launch_bounds__(256) void gcn_gemm_wmma(const float* __restrict__ x,
                                                     const float* __restrict__ W,
                                                     float* __restrict__ h, int n) {
    __shared__ float sW[(N_IN / 2) * WP_STRIDE];   // 40 KB
    for (int i = threadIdx.x; i < N_IN * N_OUT; i += blockDim.x) {
        int k = i >> 6;          // 0..127
        int c = i & 63;          // 0..63
        sW[(k >> 1) * WP_STRIDE + c * 2 + (k & 1)] = W[i];
    }
    __syncthreads();

    const int wave   = threadIdx.x >> 5;
    const int lane   = threadIdx.x & 31;
    const int rowblk = blockIdx.x * 8 + wave;
    const int m0     = rowblk * 16;
    if (m0 >= n) return;

    const int nc   = lane & 15;             // N within a 16-wide tile
    const int kgrp = (lane >> 4) << 1;      // 0 or 2: K-pair selector
    int mrow = m0 + (lane & 15);
    if (mrow >= n) mrow = n - 1;            // clamp for load; stores are guarded
    const float* xr = x + (size_t)mrow * N_IN;

    // per-lane base into packed W: pair row ((k0>>1) + lane/16), column nc
    const float* wbase = sW + (size_t)(lane >> 4) * WP_STRIDE + nc * 2;

    v8f acc0 = {}, acc1 = {}, acc2 = {}, acc3 = {};

    #pragma unroll 4
    for (int k0 = 0; k0 < N_IN; k0 += 4) {
        // A fragment: one b64 global load per lane
        v2f A = *(const v2f*)(xr + k0 + kgrp);

        // B fragments: four b64 LDS loads, immediate offsets, no shuffles
        const float* wp = wbase + (size_t)(k0 >> 1) * WP_STRIDE;
        v2f B0 = *(const v2f*)(wp +  0);
        v2f B1 = *(const v2f*)(wp + 32);
        v2f B2 = *(const v2f*)(wp + 64);
        v2f B3 = *(const v2f*)(wp + 96);

        acc0 = __builtin_amdgcn_wmma_f32_16x16x4_f32(false, A, false, B0, (short)0, acc0, false, false);
        acc1 = __builtin_amdgcn_wmma_f32_16x16x4_f32(false, A, false, B1, (short)0, acc1, false, false);
        acc2 = __builtin_amdgcn_wmma_f32_16x16x4_f32(false, A, false, B2, (short)0, acc2, false, false);
        acc3 = __builtin_amdgcn_wmma_f32_16x16x4_f32(false, A, false, B3, (short)0, acc3, false, false);
    }

    const int mbase = m0 + ((lane >> 4) << 3);   // +8 for upper half-wave
    #pragma unroll
    for (int v = 0; v < 8; ++v) {
        int r = mbase + v;
        if (r < n) {
            float* hr = h + (size_t)r * N_OUT + nc;
            hr[0]  = acc0[v];
            hr[16] = acc1[v];
            hr[32] = acc2[v];
            hr[48] = acc3[v];
        }
    }
}

// ---------------------------------------------- out = b + h * dinv^2 --------
__global__ void gcn_init_out(const float* __restrict__ h, const float* __restrict__ dinv,
                             const float* __restrict__ b, float* __restrict__ out, int n) {
    int idx = blockIdx.x * blockDim.x + threadIdx.x;
    if (idx < n * N_OUT) {
        int i = idx >> 6;
        int k = idx & 63;
        float di = dinv[i];
        out[idx] = b[k] + h[idx] * di * di;
    }
}

// ----------------------------------------------------- edge scatter ---------
// One wave32 per edge; lane handles 2 channels (b64 gather + 2 f32 atomics).
// h and out are L2-resident (25.6 MB each << 192 MB L2).
__global__ __launch_bounds__(256) void gcn_scatter(const int* __restrict__ row,
                                                   const int* __restrict__ col,
                                                   const float* __restrict__ h,
                                                   const float* __restrict__ dinv,
                                                   float* __restrict__ out, int E) {
    int gid  = blockIdx.x * blockDim.x + threadIdx.x;
    int e    = gid >> 5;
    int lane = threadIdx.x & 31;
    if (e >= E) return;

    int r = row[e];
    int c = col[e];
    float norm = dinv[r] * dinv[c];

    v2f hv = *(const v2f*)(h + (size_t)r * N_OUT + lane * 2);
    float* op = out + (size_t)c * N_OUT + lane * 2;
    atomicAdd(op + 0, hv.x * norm);
    atomicAdd(op + 1, hv.y * norm);
}

// ---------------------------------------------------------------------------
extern "C" void kernel_launch(void* const* d_in, const int* in_sizes, int n_in,
                              void* d_out, int out_size, void* d_ws, size_t ws_size,
                              hipStream_t stream) {
    const float* x  = (const float*)d_in[0];   // [N, 128]
    const int*   ei = (const int*)d_in[1];     // [2, E] flat: row then col
    const float* W  = (const float*)d_in[2];   // [128, 64]
    const float* b  = (const float*)d_in[3];   // [64]
    float* out = (float*)d_out;                // [N, 64]

    const int n = in_sizes[0] / N_IN;
    const int E = in_sizes[1] / 2;
    const int* row = ei;
    const int* col = ei + E;

    // workspace: dinv[n] then h[n*64]
    char* ws = (char*)d_ws;
    float* dinv = (float*)ws;
    size_t off = (((size_t)n * sizeof(float)) + 255) & ~(size_t)255;
    float* h = (float*)(ws + off);

    gcn_init_deg <<<(n + 255) / 256, 256, 0, stream>>>(dinv, n);
    gcn_deg_accum<<<(E + 255) / 256, 256, 0, stream>>>(col, dinv, E);
    gcn_rsqrt    <<<(n + 255) / 256, 256, 0, stream>>>(dinv, n);

    const int rowblks = (n + 15) / 16;
    gcn_gemm_wmma<<<(rowblks + 7) / 8, 256, 0, stream>>>(x, W, h, n);

    gcn_init_out <<<((n * N_OUT) + 255) / 256, 256, 0, stream>>>(h, dinv, b, out, n);

    const long long sthreads = (long long)E * 32;
    gcn_scatter  <<<(int)((sthreads + 255) / 256), 256, 0, stream>>>(row, col, h, dinv, out, E);
}